// HGCNDecoder_15564961481504
// MI455X (gfx1250) — compile-verified
//
#include <hip/hip_runtime.h>
#include <math.h>

typedef __attribute__((ext_vector_type(2))) float v2f;
typedef __attribute__((ext_vector_type(8))) float v8f;

#define EPSF      1e-7f
#define MAX_TANH  15.0f
#define MAXNORM   (1.0f - 1e-5f)   /* (1-1e-5)/sqrt(c), c = 1 */
#define DIM       64

__device__ __forceinline__ float clampf(float x, float lo, float hi) {
    return fminf(fmaxf(x, lo), hi);
}
__device__ __forceinline__ float artanh_c(float x) {
    return atanhf(clampf(x, -1.0f + 1e-5f, 1.0f - 1e-5f));
}

// ---------------------------------------------------------------------------
// hyp_bias: hb = proj(expmap0(b))   (one 64-thread block per bias vector)
// ---------------------------------------------------------------------------
__global__ __launch_bounds__(DIM)
void hyp_bias_kernel(const float* __restrict__ b, float* __restrict__ hb) {
    __shared__ float red[DIM];
    const int t = threadIdx.x;
    float v = b[t];
    red[t] = v * v;
    __syncthreads();
    for (int s = DIM / 2; s > 0; s >>= 1) {
        if (t < s) red[t] += red[t + s];
        __syncthreads();
    }
    float n  = fmaxf(sqrtf(red[0]), EPSF);
    float th = tanhf(fminf(n, MAX_TANH));
    float f  = th / n;                       // expmap0 factor
    float hn = fmaxf(fabsf(th), EPSF);       // norm of expmap0(b)
    if (hn > MAXNORM) f *= MAXNORM / hn;     // proj
    hb[t] = f * v;
}

// ---------------------------------------------------------------------------
// zero workspace (agg + deg) — grid-stride
// ---------------------------------------------------------------------------
__global__ void zero_kernel(float* __restrict__ p, long long n) {
    long long i = (long long)blockIdx.x * blockDim.x + threadIdx.x;
    long long stride = (long long)gridDim.x * blockDim.x;
    for (; i < n; i += stride) p[i] = 0.0f;
}

// ---------------------------------------------------------------------------
// HypLinear + logmap0:
//   ht = logmap0( proj( mobius_add( proj(mobius_matvec(W,x)), hb ) ) )
// Block: 256 threads = 8 waves; each wave computes a 16-row tile of x@W^T
// with V_WMMA_F32_16X16X4_F32 (16 K-steps x 4 N-tiles), results staged in
// LDS; phase 2: 128 threads do the per-row hyperbolic math.
// ---------------------------------------------------------------------------
__global__ __launch_bounds__(256)
void hyp_linear_kernel(const float* __restrict__ x,
                       const float* __restrict__ W,
                       const float* __restrict__ hb,
                       float* __restrict__ ht,
                       int nNodes) {
    __shared__ float Wl[DIM][DIM + 2];       // Wl[n][k] = W[n*64+k]
    __shared__ float mxl[128][DIM + 2];      // per-block 128x64 result tile
    __shared__ float hbs[DIM];

    const int tid     = threadIdx.x;
    const int rowBase = blockIdx.x * 128;

    for (int i = tid; i < DIM * DIM; i += 256)
        Wl[i >> 6][i & 63] = W[i];
    if (tid < DIM) hbs[tid] = hb[tid];
    __syncthreads();

    // ---- phase 1: WMMA GEMM  mx = x @ W^T -------------------------------
    const int wave = tid >> 5;
    const int lane = tid & 31;
    const int m    = lane & 15;              // row within 16-row A tile / N within B tile
    const int kb   = (lane >> 4) << 1;       // K sub-offset: lanes 0-15 -> K0,K1; 16-31 -> K2,K3

    int gRow = rowBase + wave * 16 + m;
    if (gRow >= nNodes) gRow = nNodes - 1;   // clamp (EXEC must stay all-ones for WMMA)
    const float* xr = x + (long long)gRow * DIM;

    v8f acc0 = {}, acc1 = {}, acc2 = {}, acc3 = {};
    #pragma unroll
    for (int k0 = 0; k0 < DIM; k0 += 4) {
        v2f a;
        a.x = xr[k0 + kb];
        a.y = xr[k0 + kb + 1];
        v2f b0, b1, b2, b3;
        b0.x = Wl[ 0 + m][k0 + kb]; b0.y = Wl[ 0 + m][k0 + kb + 1];
        b1.x = Wl[16 + m][k0 + kb]; b1.y = Wl[16 + m][k0 + kb + 1];
        b2.x = Wl[32 + m][k0 + kb]; b2.y = Wl[32 + m][k0 + kb + 1];
        b3.x = Wl[48 + m][k0 + kb]; b3.y = Wl[48 + m][k0 + kb + 1];
        acc0 = __builtin_amdgcn_wmma_f32_16x16x4_f32(false, a, false, b0, (short)0, acc0, false, false);
        acc1 = __builtin_amdgcn_wmma_f32_16x16x4_f32(false, a, false, b1, (short)0, acc1, false, false);
        acc2 = __builtin_amdgcn_wmma_f32_16x16x4_f32(false, a, false, b2, (short)0, acc2, false, false);
        acc3 = __builtin_amdgcn_wmma_f32_16x16x4_f32(false, a, false, b3, (short)0, acc3, false, false);
    }
    // C/D layout: VGPR v -> M = v + 8*(lane>>4), N = lane&15
    const int rOff = wave * 16 + ((lane >> 4) << 3);
    #pragma unroll
    for (int v = 0; v < 8; ++v) {
        mxl[rOff + v][ 0 + m] = acc0[v];
        mxl[rOff + v][16 + m] = acc1[v];
        mxl[rOff + v][32 + m] = acc2[v];
        mxl[rOff + v][48 + m] = acc3[v];
    }
    __syncthreads();

    // ---- phase 2: per-row hyperbolic math -------------------------------
    if (tid < 128) {
        const int g = rowBase + tid;
        if (g < nNodes) {
            const float* xg  = x + (long long)g * DIM;
            float*       row = mxl[tid];

            float xn2 = 0.f, mxn2 = 0.f;
            for (int j = 0; j < DIM; ++j) {
                float xv = xg[j];  xn2  += xv * xv;
                float mv = row[j]; mxn2 += mv * mv;
            }
            float xn  = fmaxf(sqrtf(xn2),  EPSF);
            float mxn = fmaxf(sqrtf(mxn2), EPSF);
            float t   = tanhf(clampf(mxn / xn * artanh_c(xn), -MAX_TANH, MAX_TANH));
            float s   = t / mxn;                 // mobius_matvec scale on mx
            float hn  = fmaxf(fabsf(t), EPSF);   // resulting norm
            if (hn > MAXNORM) s *= MAXNORM / hn; // proj

            // mobius_add(h, hb)
            float x2 = 0.f, y2 = 0.f, xy = 0.f;
            for (int j = 0; j < DIM; ++j) {
                float hv = s * row[j]; row[j] = hv;
                float pv = hbs[j];
                x2 += hv * hv; y2 += pv * pv; xy += hv * pv;
            }
            float ca  = 1.f + 2.f * xy + y2;
            float cb  = 1.f - x2;
            float den = fmaxf(1.f + 2.f * xy + x2 * y2, EPSF);
            float n2s = 0.f;
            for (int j = 0; j < DIM; ++j) {
                float hv = (ca * row[j] + cb * hbs[j]) / den;
                row[j] = hv; n2s += hv * hv;
            }
            // proj + logmap0 (fused scalar factor)
            float n  = fmaxf(sqrtf(n2s), EPSF);
            float pr = (n > MAXNORM) ? (MAXNORM / n) : 1.0f;
            float nn = fmaxf(n * pr, EPSF);
            float lf = artanh_c(nn) / nn * pr;
            float* dst = ht + (long long)g * DIM;
            for (int j = 0; j < DIM; ++j) dst[j] = lf * row[j];
        }
    }
}

// ---------------------------------------------------------------------------
// Edge aggregation: 16 lanes per edge, float4 row gather + f32 atomics into
// the L2-resident agg/deg buffers.
// ---------------------------------------------------------------------------
__global__ __launch_bounds__(256)
void hyp_agg_kernel(const long long* __restrict__ edges,
                    const float* __restrict__ edge_mask,
                    const float* __restrict__ ht,
                    float* __restrict__ agg,
                    float* __restrict__ deg,
                    int nEdges) {
    const long long t = (long long)blockIdx.x * blockDim.x + threadIdx.x;
    const long long e = t >> 4;
    if (e >= nEdges) return;
    const int part = (int)(t & 15);

    const long long row = edges[e];
    const long long col = edges[(long long)nEdges + e];
    const float     msk = edge_mask[e];

    const float4 v = ((const float4*)(ht + col * DIM))[part];
    float* dst = agg + row * DIM + part * 4;
    atomicAdd(dst + 0, v.x * msk);
    atomicAdd(dst + 1, v.y * msk);
    atomicAdd(dst + 2, v.z * msk);
    atomicAdd(dst + 3, v.w * msk);
    if (part == 0) atomicAdd(deg + row, msk);
}

// ---------------------------------------------------------------------------
// HypAgg tail + HypAct:
//   out = proj(expmap0(relu(logmap0(proj(expmap0(agg/deg)))))) * node_mask
// One thread per node; 64-wide row kept in registers (unrolled).
// ---------------------------------------------------------------------------
__global__ __launch_bounds__(256)
void hyp_finalize_kernel(const float* __restrict__ agg,
                         const float* __restrict__ deg,
                         const float* __restrict__ node_mask,
                         float* __restrict__ out,
                         int nNodes) {
    const int i = blockIdx.x * blockDim.x + threadIdx.x;
    if (i >= nNodes) return;

    const float* a   = agg + (long long)i * DIM;
    const float  inv = 1.0f / fmaxf(deg[i], 1.0f);

    float u[DIM];
    float n2 = 0.f;
    #pragma unroll
    for (int j = 0; j < DIM; ++j) { u[j] = a[j] * inv; n2 += u[j] * u[j]; }

    // h = proj(expmap0(u))
    float n  = fmaxf(sqrtf(n2), EPSF);
    float th = tanhf(fminf(n, MAX_TANH));
    float f  = th / n;
    float hn = fmaxf(fabsf(th), EPSF);
    float pr = (hn > MAXNORM) ? (MAXNORM / hn) : 1.0f;
    f  *= pr;
    hn *= pr;
    // xt = relu(logmap0(h))
    const float lf = artanh_c(hn) / hn;
    float nx2 = 0.f;
    #pragma unroll
    for (int j = 0; j < DIM; ++j) {
        float xt = fmaxf(lf * f * u[j], 0.0f);
        u[j] = xt; nx2 += xt * xt;
    }
    // out = proj(expmap0(xt)) * node_mask
    float nx  = fmaxf(sqrtf(nx2), EPSF);
    float th2 = tanhf(fminf(nx, MAX_TANH));
    float f2  = th2 / nx;
    float hn2 = fmaxf(fabsf(th2), EPSF);
    if (hn2 > MAXNORM) f2 *= MAXNORM / hn2;
    f2 *= node_mask[i];

    float* o = out + (long long)i * DIM;
    #pragma unroll
    for (int j = 0; j < DIM; ++j) o[j] = f2 * u[j];
}

// ---------------------------------------------------------------------------
extern "C" void kernel_launch(void* const* d_in, const int* in_sizes, int n_in,
                              void* d_out, int out_size, void* d_ws, size_t ws_size,
                              hipStream_t stream) {
    const float*     h         = (const float*)d_in[0];
    /* d_in[1] = distances (unused by reference) */
    const long long* edges     = (const long long*)d_in[2];  // int64 [2,E]
    const float*     node_mask = (const float*)d_in[3];
    const float*     edge_mask = (const float*)d_in[4];
    const float*     W0        = (const float*)d_in[5];
    const float*     b0        = (const float*)d_in[6];
    const float*     W1        = (const float*)d_in[7];
    const float*     b1        = (const float*)d_in[8];

    const int N = in_sizes[0] / DIM;     // 50000
    const int E = in_sizes[4];           // edge_mask flat count = 1200000

    // workspace layout (f32)
    float* ws  = (float*)d_ws;
    float* ht  = ws;                           // N*DIM
    float* agg = ht  + (size_t)N * DIM;        // N*DIM
    float* x1  = agg + (size_t)N * DIM;        // N*DIM
    float* dg  = x1  + (size_t)N * DIM;        // N
    float* hb0 = dg  + N;                      // DIM
    float* hb1 = hb0 + DIM;                    // DIM

    hyp_bias_kernel<<<1, DIM, 0, stream>>>(b0, hb0);
    hyp_bias_kernel<<<1, DIM, 0, stream>>>(b1, hb1);

    const int grdLin  = (N + 127) / 128;
    const int grdAgg  = (int)(((long long)E * 16 + 255) / 256);
    const int grdFin  = (N + 255) / 256;
    const long long zn = (long long)N * DIM + N;   // agg + deg element count (x1 lies between them)

    // ---- layer 1 ----
    hyp_linear_kernel<<<grdLin, 256, 0, stream>>>(h, W0, hb0, ht, N);
    zero_kernel<<<2048, 256, 0, stream>>>(agg, 0);           // no-op guard (keeps capture shape)
    zero_kernel<<<2048, 256, 0, stream>>>(agg, (long long)N * DIM);
    zero_kernel<<<256, 256, 0, stream>>>(dg, (long long)N);
    hyp_agg_kernel<<<grdAgg, 256, 0, stream>>>(edges, edge_mask, ht, agg, dg, E);
    hyp_finalize_kernel<<<grdFin, 256, 0, stream>>>(agg, dg, node_mask, x1, N);

    // ---- layer 2 ----
    hyp_linear_kernel<<<grdLin, 256, 0, stream>>>(x1, W1, hb1, ht, N);
    zero_kernel<<<2048, 256, 0, stream>>>(agg, (long long)N * DIM);
    zero_kernel<<<256, 256, 0, stream>>>(dg, (long long)N);
    hyp_agg_kernel<<<grdAgg, 256, 0, stream>>>(edges, edge_mask, ht, agg, dg, E);
    hyp_finalize_kernel<<<grdFin, 256, 0, stream>>>(agg, dg, node_mask, (float*)d_out, N);
}